// DRR_30434138259702
// MI455X (gfx1250) — compile-verified
//
#include <hip/hip_runtime.h>
#include <hip/hip_bf16.h>

#define NXD 192
#define NVOL (192 * 192 * 192)
#define HD 192
#define WD 192
#define NSUB ((HD * WD) / 2)
#define BD 2
#define SDRC 300.0f
#define DELC 2.0f
#define SXY (192 * 192)
#define NSEG 4   // α-range split per ray (lanes 4k..4k+3 share a ray)

// ---- order-preserving float <-> uint keys (for atomic min/max) ----
__device__ __forceinline__ unsigned fkey(float f) {
    unsigned u = __float_as_uint(f);
    return u ^ ((u >> 31) ? 0xFFFFFFFFu : 0x80000000u);
}
__device__ __forceinline__ float fdec(unsigned k) {
    unsigned u = k ^ ((k >> 31) ? 0x80000000u : 0xFFFFFFFFu);
    return __uint_as_float(u);
}

// ---- kernel 1: zero output, re-init reduction slots (every launch) ----
__global__ __launch_bounds__(256) void zero_init(float* __restrict__ out, int n,
                                                 unsigned* __restrict__ slots) {
    int i = blockIdx.x * blockDim.x + threadIdx.x;
    if (i < n) out[i] = 0.0f;
    if (i == 0) {
        slots[0] = fkey(__builtin_inff());   // soft_min  (min-reduce)
        slots[1] = fkey(-__builtin_inff());  // max over v > -800 (max-reduce)
    }
}

// ---- kernel 2: volume reductions via async global->LDS (CDNA5 path) ----
__global__ __launch_bounds__(256) void reduce_minmax(const float* __restrict__ vol,
                                                     unsigned* __restrict__ slots,
                                                     int nvec4) {
    // single LDS object -> starts at LDS offset 0; partition manually
    __shared__ unsigned char smem[256 * 16 + 2 * 256 * 4];
    float* redA = (float*)(smem + 256 * 16);
    float* redB = redA + 256;

    const unsigned tid = threadIdx.x;
    const unsigned ldsoff = tid * 16u;  // this lane's b128 staging slot

    float smin = __builtin_inff();
    float rmax = -__builtin_inff();

    long long stride = (long long)gridDim.x * blockDim.x;
    for (long long i = (long long)blockIdx.x * blockDim.x + tid; i < nvec4; i += stride) {
        unsigned long long ga = (unsigned long long)(vol + 4 * i);
        // CDNA5 async copy: global -> LDS, 16 bytes per lane
        asm volatile("global_load_async_to_lds_b128 %0, %1, off"
                     :: "v"(ldsoff), "v"(ga) : "memory");
        asm volatile("s_wait_asynccnt 0x0" ::: "memory");
        volatile float* sp = (volatile float*)(smem + ldsoff);
#pragma unroll
        for (int c = 0; c < 4; ++c) {
            float v = sp[c];
            if (v > -800.0f) {
                rmax = fmaxf(rmax, v);
                if (v <= 350.0f) smin = fminf(smin, v);
            }
        }
    }

    redA[tid] = smin;
    redB[tid] = rmax;
    __syncthreads();
    for (int s = 128; s > 0; s >>= 1) {
        if (tid < (unsigned)s) {
            redA[tid] = fminf(redA[tid], redA[tid + s]);
            redB[tid] = fmaxf(redB[tid], redB[tid + s]);
        }
        __syncthreads();
    }
    if (tid == 0) {
        atomicMin(&slots[0], fkey(redA[0]));
        atomicMax(&slots[1], fkey(redB[0]));
    }
}

// ---- kernel 3: incremental Siddon DDA; NSEG lanes cooperate per ray ----
__global__ __launch_bounds__(256) void drr_ray(const float* __restrict__ vol,
                                               const float* __restrict__ rot,
                                               const float* __restrict__ trs,
                                               const int* __restrict__ sub,
                                               const unsigned* __restrict__ slots,
                                               float* __restrict__ out) {
    int gid = blockIdx.x * blockDim.x + threadIdx.x;
    if (gid >= BD * NSUB * NSEG) return;
    int seg = gid & (NSEG - 1);
    int rid = gid >> 2;            // ray id
    int b = rid / NSUB;
    int n = rid - b * NSUB;
    int p = sub[n];
    int row = p / WD;
    int col = p - row * WD;

    // normalization constants (BAM == 1: >350 branch is identity)
    float smv  = fdec(slots[0]);
    float mxr  = fdec(slots[1]);
    float dmin = smv;
    float dmax = fmaxf(smv, mxr);
    float scale = (dmax > dmin) ? 1.0f / (dmax - dmin) : 0.0f;

    // rotation R = Rz(r0) Ry(r1) Rx(r2)
    float r0 = rot[b * 3 + 0], r1 = rot[b * 3 + 1], r2 = rot[b * 3 + 2];
    float cz = cosf(r0), sz = sinf(r0);
    float cy = cosf(r1), sy = sinf(r1);
    float cx = cosf(r2), sx = sinf(r2);
    float R00 = cz * cy, R01 = cz * sy * sx - sz * cx, R02 = cz * sy * cx + sz * sx;
    float R10 = sz * cy, R11 = sz * sy * sx + cz * cx, R12 = sz * sy * cx - cz * sx;
    float R20 = -sy,     R21 = cy * sx,                R22 = cy * cx;

    float tx = trs[b * 3 + 0] + 96.0f;
    float ty = trs[b * 3 + 1] + 96.0f;
    float tz = trs[b * 3 + 2] + 96.0f;

    // source = R @ (0,0,SDR) + t
    float sxr = R02 * SDRC + tx;
    float syr = R12 * SDRC + ty;
    float szr = R22 * SDRC + tz;
    // target = R @ (gx,gy,-SDR) + t
    float gx = ((float)col - 95.5f) * DELC;
    float gy = ((float)row - 95.5f) * DELC;
    float txr = R00 * gx + R01 * gy - R02 * SDRC + tx;
    float tyr = R10 * gx + R11 * gy - R12 * SDRC + ty;
    float tzr = R20 * gx + R21 * gy - R22 * SDRC + tz;

    float dx = txr - sxr, dy = tyr - syr, dz = tzr - szr;
    float rlen = sqrtf(dx * dx + dy * dy + dz * dz);
    float dxs = (dx == 0.0f) ? 1e-9f : dx;   // matches reference sdd_safe
    float dys = (dy == 0.0f) ? 1e-9f : dy;
    float dzs = (dz == 0.0f) ? 1e-9f : dz;
    float inx = 1.0f / dxs, iny = 1.0f / dys, inz = 1.0f / dzs;

    float a0x = (0.0f - sxr) * inx, a1x = (192.0f - sxr) * inx;
    float a0y = (0.0f - syr) * iny, a1y = (192.0f - syr) * iny;
    float a0z = (0.0f - szr) * inz, a1z = (192.0f - szr) * inz;
    float amin = fmaxf(fmaxf(fminf(a0x, a1x), fminf(a0y, a1y)), fminf(a0z, a1z));
    amin = fmaxf(amin, 0.0f);
    float amax = fminf(fminf(fmaxf(a0x, a1x), fmaxf(a0y, a1y)), fmaxf(a0z, a1z));
    amax = fminf(amax, 1.0f);
    amax = fmaxf(amax, amin);

    // this lane's α sub-range; identical formulas on adjacent lanes ->
    // seg k's end bit-matches seg k+1's start (no gap/overlap)
    float q = 0.25f * (amax - amin);
    float aS = amin + (float)seg * q;
    float aE = (seg == NSEG - 1) ? amax : amin + (float)(seg + 1) * q;

    float acc = 0.0f;
    if (aE > aS) {
        float px = sxr + aS * dx;
        float py = syr + aS * dy;
        float pz = szr + aS * dz;
        int ix = min(max((int)floorf(px), 0), 191);
        int iy = min(max((int)floorf(py), 0), 191);
        int iz = min(max((int)floorf(pz), 0), 191);
        int stx = (dxs > 0.0f) ? 1 : -1;
        int sty = (dys > 0.0f) ? 1 : -1;
        int stz = (dzs > 0.0f) ? 1 : -1;
        float dax = fabsf(inx), day = fabsf(iny), daz = fabsf(inz);
        float axn = ((float)(stx > 0 ? ix + 1 : ix) - sxr) * inx;
        float ayn = ((float)(sty > 0 ? iy + 1 : iy) - syr) * iny;
        float azn = ((float)(stz > 0 ? iz + 1 : iz) - szr) * inz;

        // warm L2/L0 for the entry voxel (gfx1250 global_prefetch_b8)
        __builtin_prefetch(&vol[(191 - ix) * SXY + iy * 192 + iz], 0, 3);

        float alpha = aS;
        for (int it = 0; it < 640; ++it) {
            float an = fminf(fminf(axn, ayn), fminf(azn, aE));
            // flip folded into fetch; density transform on the fly.
            // loaded value feeds only 'acc' -> traversal chain never waits on it
            float v = vol[(191 - ix) * SXY + iy * 192 + iz];
            float dens = (((v <= -800.0f) ? smv : v) - dmin) * scale;
            acc += fmaxf(an - alpha, 0.0f) * dens;
            alpha = an;
            if (an >= aE) break;
            if (axn <= ayn && axn <= azn) {
                ix += stx; axn += dax; if ((unsigned)ix > 191u) break;
            } else if (ayn <= azn) {
                iy += sty; ayn += day; if ((unsigned)iy > 191u) break;
            } else {
                iz += stz; azn += daz; if ((unsigned)iz > 191u) break;
            }
        }
    }

    // deterministic fixed-shape combine of the 4 segment partials
    // (lane pairing is fixed; IEEE add is bitwise-commutative)
    acc += __shfl_xor(acc, 1);
    acc += __shfl_xor(acc, 2);
    if (seg == 0) out[b * (HD * WD) + p] = acc * rlen;
}

extern "C" void kernel_launch(void* const* d_in, const int* in_sizes, int n_in,
                              void* d_out, int out_size, void* d_ws, size_t ws_size,
                              hipStream_t stream) {
    const float* vol = (const float*)d_in[0];       // [192,192,192] f32
    const float* rot = (const float*)d_in[1];       // [2,3] f32
    const float* trs = (const float*)d_in[2];       // [2,3] f32
    const int*   sub = (const int*)d_in[3];         // [18432] i32
    float* out = (float*)d_out;                     // [2,1,192,192] f32
    unsigned* slots = (unsigned*)d_ws;              // 2 reduction slots

    int n_out = BD * HD * WD;
    if (n_out > out_size) n_out = out_size;

    zero_init<<<(n_out + 255) / 256, 256, 0, stream>>>(out, n_out, slots);
    reduce_minmax<<<1024, 256, 0, stream>>>(vol, slots, NVOL / 4);
    int nthreads = BD * NSUB * NSEG;                 // 147456 -> 4608 wave32s
    drr_ray<<<(nthreads + 255) / 256, 256, 0, stream>>>(vol, rot, trs, sub, slots, out);
}